// TopLoss_62861141344730
// MI455X (gfx1250) — compile-verified
//
#include <hip/hip_runtime.h>
#include <stdint.h>

// Problem constants (match reference)
#define HDIM     400
#define CS       32
#define INNER    30          // CS - 2*PAD
#define NPX      1024        // CS*CS
#define NTHREADS 256
#define NSWEEPS  32          // B(4) * NUM_CHUNKS(2) * 4 sweeps
#define NCHUNKS  8           // B * NUM_CHUNKS

#if defined(__gfx1250__)
#define USE_CDNA5_ASYNC 1
#else
#define USE_CDNA5_ASYNC 0
#endif

typedef unsigned int u32x4 __attribute__((ext_vector_type(4)));
typedef unsigned int u32x8 __attribute__((ext_vector_type(8)));

// Map float -> uint with identical total order (no NaNs in this workload).
__device__ inline unsigned f2o(float f) {
  unsigned u = __float_as_uint(f);
  return (u & 0x80000000u) ? ~u : (u | 0x80000000u);
}

// Union-find find with path halving. Compression is output-equivalent to the
// reference's plain chain walk: roots and merge decisions are unchanged.
__device__ inline int uf_find(int* parent, int p) {
  while (parent[p] != p) {
    parent[p] = parent[parent[p]];
    p = parent[p];
  }
  return p;
}

// In-LDS bitonic sort, ascending, NPX 64-bit keys, NTHREADS threads.
__device__ inline void bitonic_sort(unsigned long long* keys, int tid) {
  for (int size = 2; size <= NPX; size <<= 1) {
    for (int stride = size >> 1; stride > 0; stride >>= 1) {
      __syncthreads();
      for (int p = tid; p < NPX / 2; p += NTHREADS) {
        int i = ((p & ~(stride - 1)) << 1) | (p & (stride - 1));
        int j = i | stride;
        bool desc = (i & size) != 0;
        unsigned long long a = keys[i];
        unsigned long long b = keys[j];
        bool doswap = desc ? (a < b) : (a > b);
        if (doswap) { keys[i] = b; keys[j] = a; }
      }
    }
  }
  __syncthreads();
}

// One workgroup per sweep: sweep = ((b*2 + c)*4 + t)
//   t=0: h0(-target)  t=1: h0(+target)  t=2: h0(-input)  t=3: h0(+input)
// Target sweeps write count(lengths>0) to ws_prior[chunk*2+t].
// Input sweeps write ascending-sorted lengths to ws_sorted[sweep*NPX ..].
__global__ __launch_bounds__(NTHREADS)
void topo_sweep_kernel(const float* __restrict__ input,
                       const float* __restrict__ target,
                       const int*   __restrict__ offsets,
                       float* __restrict__ ws_sorted,
                       int*   __restrict__ ws_prior) {
  __shared__ float val[NPX];           // padded chunk values (reused as slen out)
  __shared__ float lengths[NPX];
  __shared__ int parent[NPX];
  __shared__ unsigned long long keys[NPX];
  __shared__ float stage[INNER * INNER];  // compact 30x30 TDM landing tile
  __shared__ int s_cnt;

  const int tid   = threadIdx.x;
  const int sweep = blockIdx.x;
  const int t     = sweep & 3;
  const int chunk = sweep >> 2;              // = b*NUM_CHUNKS + c
  const int b     = chunk >> 1;              // NUM_CHUNKS == 2
  const float sign = (t == 0 || t == 2) ? -1.0f : 1.0f;
  const float* img = (t < 2) ? target : input;
  const int off0 = offsets[chunk * 2 + 0];
  const int off1 = offsets[chunk * 2 + 1];
  const float* base = img + (size_t)b * HDIM * HDIM + (size_t)off0 * HDIM + off1;

  // ---- Tile extraction: Tensor Data Mover DMA (global -> LDS) ------------
#if USE_CDNA5_ASYNC
  if (tid < 32) {  // one wave issues the TDM op (TDM ignores EXEC; per-wave)
    unsigned long long ga = (unsigned long long)(const void*)base;
    unsigned ldsa = (unsigned)(unsigned long long)(void*)&stage[0];
    // D# group 0: count=1 (valid, load, no gather), lds_addr, global_addr, type=2
    u32x4 g0 = { 1u, ldsa, (unsigned)ga,
                 ((unsigned)((ga >> 32) & 0x01FFFFFFull)) | 0x80000000u };
    // D# group 1: data_size=4B; tensor 400x400 stride 400; tile 30x30; no pad
    u32x8 g1 = { 0x00020000u,                 // wg_mask=0 | data_size=2 (4B)
                 ((unsigned)HDIM) << 16,      // tensor_dim0[15:0]=400
                 ((unsigned)HDIM) << 16,      // dim0 hi=0 | tensor_dim1[15:0]=400
                 ((unsigned)INNER) << 16,     // dim1 hi=0 | tile_dim0=30
                 (unsigned)INNER,             // tile_dim1=30 | tile_dim2=0
                 (unsigned)HDIM,              // tensor_dim0_stride[31:0]=400
                 0u, 0u };                    // stride hi=0 | tensor_dim1_stride=0
    asm volatile("tensor_load_to_lds %0, %1" :: "s"(g0), "s"(g1) : "memory");
    __builtin_amdgcn_s_wait_tensorcnt(0);
  }
  __syncthreads();
#endif

  // ---- Build padded/negated chunk in LDS; init union-find state ----------
  for (int i = tid; i < NPX; i += NTHREADS) {
    int r = i >> 5, c = i & 31;
    bool interior = (r >= 1) && (r <= INNER) && (c >= 1) && (c <= INNER);
#if USE_CDNA5_ASYNC
    float f = interior ? stage[(r - 1) * INNER + (c - 1)] : 1.0f;
#else
    float f = interior ? base[(size_t)(r - 1) * HDIM + (c - 1)] : 1.0f;
#endif
    val[i]     = sign * f;
    parent[i]  = -1;
    lengths[i] = 0.0f;
  }
  __syncthreads();

  // ---- Stable sort by (value, index): replicates jnp.argsort order -------
  for (int i = tid; i < NPX; i += NTHREADS)
    keys[i] = ((unsigned long long)f2o(val[i]) << 32) | (unsigned)i;
  bitonic_sort(keys, tid);

  // ---- Sequential elder-rule union-find sweep (single lane, all-LDS) -----
  if (tid == 0) {
    const int di[6] = {-1, 1, 0, 0, 1, -1};
    const int dj[6] = { 0, 0,-1, 1, 1, -1};
    for (int s = 0; s < NPX; ++s) {
      int u    = (int)(keys[s] & 0xFFFFFFFFu);
      float vu = val[u];
      parent[u] = u;
      int ur = u >> 5, uc = u & 31;
      for (int d = 0; d < 6; ++d) {
        int nr = ur + di[d], nc = uc + dj[d];
        if (nr < 0 || nr >= CS || nc < 0 || nc >= CS) continue;
        int nb = (nr << 5) | nc;
        if (parent[nb] < 0) continue;          // neighbor not inserted yet
        int ru = uf_find(parent, u);
        int rn = uf_find(parent, nb);
        if (ru == rn) continue;
        float bu = val[ru], bn = val[rn];      // birth[root] == val[root]
        bool u_elder = (bu < bn) || ((bu == bn) && (ru <= rn));
        int elder = u_elder ? ru : rn;
        int young = u_elder ? rn : ru;
        lengths[young] = vu - val[young];      // younger bar dies now
        parent[young]  = elder;
      }
    }
  }
  __syncthreads();

  // ---- Emit per-sweep result ---------------------------------------------
  if (t < 2) {
    if (tid == 0) s_cnt = 0;
    __syncthreads();
    int local = 0;
    for (int i = tid; i < NPX; i += NTHREADS) local += (lengths[i] > 0.0f) ? 1 : 0;
    atomicAdd(&s_cnt, local);
    __syncthreads();
    if (tid == 0) ws_prior[chunk * 2 + t] = s_cnt;
  } else {
    for (int i = tid; i < NPX; i += NTHREADS)
      keys[i] = ((unsigned long long)f2o(lengths[i]) << 32) | (unsigned)i;
    bitonic_sort(keys, tid);
#if USE_CDNA5_ASYNC
    // Stage sorted lengths in LDS (reuse val[], dead after the sweep), then
    // DMA them out with async LDS->global stores (ASYNCcnt-tracked).
    for (int i = tid; i < NPX; i += NTHREADS)
      val[i] = lengths[(int)(keys[i] & 0xFFFFFFFFu)];
    asm volatile("s_wait_dscnt 0" ::: "memory");   // LDS writes visible to DMA
    for (int i = tid; i < NPX; i += NTHREADS) {
      unsigned long long ga =
          (unsigned long long)(const void*)(ws_sorted + (size_t)sweep * NPX + i);
      unsigned lo = (unsigned)(unsigned long long)(void*)&val[i];
      asm volatile("global_store_async_from_lds_b32 %0, %1, off"
                   :: "v"(ga), "v"(lo) : "memory");
    }
    asm volatile("s_wait_asynccnt 0" ::: "memory");
#else
    for (int i = tid; i < NPX; i += NTHREADS)
      ws_sorted[(size_t)sweep * NPX + i] = lengths[(int)(keys[i] & 0xFFFFFFFFu)];
#endif
  }
}

// One block: combine priors + sorted input bars into the scalar mean loss.
__global__ __launch_bounds__(NTHREADS)
void topo_reduce_kernel(const float* __restrict__ ws_sorted,
                        const int*   __restrict__ ws_prior,
                        float* __restrict__ out) {
  __shared__ float red[NTHREADS];
  int tid = threadIdx.x;
  float total = 0.0f;
  for (int chunk = 0; chunk < NCHUNKS; ++chunk) {
    int prior0 = ws_prior[chunk * 2 + 0] + 1;  // +1 for essential dim-0 bar
    int prior1 = ws_prior[chunk * 2 + 1];
    int keep0 = NPX - prior0; if (keep0 < 0) keep0 = 0;
    int keep1 = NPX - prior1; if (keep1 < 0) keep1 = 0;
    const float* a0 = ws_sorted + (size_t)(chunk * 4 + 2) * NPX;  // lx0 ascending
    const float* a1 = ws_sorted + (size_t)(chunk * 4 + 3) * NPX;  // lx1 ascending
    float local = 0.0f;
    for (int i = tid; i < keep0; i += NTHREADS) local += a0[i];
    for (int i = tid; i < keep1; i += NTHREADS) local += a1[i];
    red[tid] = local;
    __syncthreads();
    for (int s = NTHREADS / 2; s > 0; s >>= 1) {
      if (tid < s) red[tid] += red[tid + s];
      __syncthreads();
    }
    if (tid == 0) total += red[0];
    __syncthreads();
  }
  if (tid == 0) out[0] = total * (1.0f / (float)NCHUNKS);
}

extern "C" void kernel_launch(void* const* d_in, const int* in_sizes, int n_in,
                              void* d_out, int out_size, void* d_ws, size_t ws_size,
                              hipStream_t stream) {
  (void)in_sizes; (void)n_in; (void)out_size; (void)ws_size;
  const float* input  = (const float*)d_in[0];
  const float* target = (const float*)d_in[1];
  const int*   offs   = (const int*)d_in[2];
  float* ws_sorted = (float*)d_ws;                                    // 32*1024 f32
  int*   ws_prior  = (int*)((char*)d_ws + (size_t)NSWEEPS * NPX * 4); // 16 i32

  topo_sweep_kernel<<<NSWEEPS, NTHREADS, 0, stream>>>(input, target, offs,
                                                      ws_sorted, ws_prior);
  topo_reduce_kernel<<<1, NTHREADS, 0, stream>>>(ws_sorted, ws_prior,
                                                 (float*)d_out);
}